// ChamferLoss_Brute_NoSigma_30992484008032
// MI455X (gfx1250) — compile-verified
//
#include <hip/hip_runtime.h>

// Problem constants fixed by the harness reference: [8, 3, 4096] point clouds.
#define B_  8
#define C_  3
#define M_  4096
#define N_  4096
#define FINF_BITS 0x7F800000u   // +inf (all sq distances are clamped >= 0)

typedef float v2f __attribute__((ext_vector_type(2)));
typedef float v8f __attribute__((ext_vector_type(8)));

// ---------------------------------------------------------------------------
// Kernel 1: precompute squared norms s2[b,m], d2[b,n]; init bmin to +inf.
// (M_ == N_ so one pass covers both arrays.)
// ---------------------------------------------------------------------------
__global__ void chamfer_init_kernel(const float* __restrict__ src,
                                    const float* __restrict__ dst,
                                    float* __restrict__ s2,
                                    float* __restrict__ d2,
                                    unsigned* __restrict__ bmin) {
    int i = blockIdx.x * blockDim.x + threadIdx.x;
    if (i >= B_ * M_) return;
    int b = i / M_;
    int m = i - b * M_;
    float sx = src[(b * 3 + 0) * M_ + m];
    float sy = src[(b * 3 + 1) * M_ + m];
    float sz = src[(b * 3 + 2) * M_ + m];
    s2[i] = sx * sx + sy * sy + sz * sz;
    float dx = dst[(b * 3 + 0) * N_ + m];
    float dy = dst[(b * 3 + 1) * N_ + m];
    float dz = dst[(b * 3 + 2) * N_ + m];
    d2[i] = dx * dx + dy * dy + dz * dz;
    bmin[i] = FINF_BITS;
}

// ---------------------------------------------------------------------------
// Kernel 2: main tiled distance kernel.
// grid = (M_/128, B_), block = 256 (8 wave32s). Each wave owns 16 M-rows and
// sweeps all N in 16-wide tiles via V_WMMA_F32_16X16X4_F32 (K=3 padded to 4).
// Fully convergent hot loop: no exec-mask changes, no cross-lane shuffles.
// ---------------------------------------------------------------------------
__global__ __launch_bounds__(256) void chamfer_tile_kernel(
    const float* __restrict__ src, const float* __restrict__ dst,
    const float* __restrict__ s2g, const float* __restrict__ d2g,
    float* __restrict__ fmin, unsigned* __restrict__ bming) {

    __shared__ unsigned sBmin[N_];   // 16 KB: per-workgroup column-min (uint bits)

    const int b    = blockIdx.y;
    const int m0   = blockIdx.x * 128;        // 8 waves * 16 rows
    const int tid  = threadIdx.x;
    const int lane = tid & 31;
    const int wave = tid >> 5;
    const int hi   = lane >> 4;               // 0: lanes 0-15, 1: lanes 16-31
    const int l15  = lane & 15;
    const float ymask = hi ? 0.0f : 1.0f;     // zeroes K=3 pad row branchlessly

    for (int n = tid; n < N_; n += 256) sBmin[n] = FINF_BITS;
    __syncthreads();

    // ---- A tile (16x4): lane layout per ISA 7.12.2 (32-bit A, 16x4) ----
    // low half lanes: (x, y) = K0,K1 rows; high half lanes: (z, 0) = K2,K3.
    const int mrow = m0 + wave * 16 + l15;
    v2f a;
    a.x = src[(b * 3 + (hi ? 2 : 0)) * M_ + mrow];
    a.y = src[(b * 3 + 1) * M_ + mrow] * ymask;

    // s2 values for the 8 rows this lane's D registers correspond to.
    float s2v[8];
#pragma unroll
    for (int r = 0; r < 8; ++r)
        s2v[r] = s2g[b * M_ + m0 + wave * 16 + r + 8 * hi];

    float fmacc[8];
#pragma unroll
    for (int r = 0; r < 8; ++r) fmacc[r] = __builtin_inff();

    const float* dstA = dst + (b * 3 + (hi ? 2 : 0)) * N_ + l15;
    const float* dstB = dst + (b * 3 + 1) * N_ + l15;
    const float* d2b  = d2g + b * N_ + l15;

#pragma unroll 4
    for (int nt = 0; nt < N_; nt += 16) {
        // ---- B tile (4x16): unconditional loads, pad row zeroed by multiply ----
        const float bx  = dstA[nt];
        const float by  = dstB[nt];
        const float d2v = d2b[nt];
        v2f bb;
        bb.x = bx;
        bb.y = by * ymask;

        v8f c = {0.f, 0.f, 0.f, 0.f, 0.f, 0.f, 0.f, 0.f};
        // D = A(16x4) * B(4x16) + 0  -> 16x16 cross-products
        v8f d = __builtin_amdgcn_wmma_f32_16x16x4_f32(
            /*neg_a=*/false, a, /*neg_b=*/false, bb,
            /*c_mod=*/(short)0, c, /*reuse_a=*/false, /*reuse_b=*/false);

        float colmin = __builtin_inff();
#pragma unroll
        for (int r = 0; r < 8; ++r) {
            float sq = __builtin_fmaf(-2.0f, d[r], s2v[r] + d2v);
            sq = fmaxf(sq, 0.0f);
            fmacc[r] = fminf(fmacc[r], sq);
            colmin   = fminf(colmin, sq);
        }
        // Lanes l and l+16 hold partial column mins (rows 0-7 vs 8-15) for the
        // SAME column n = nt+l15; let the LDS atomic unit merge them — no
        // shuffle, no exec-mask divergence in the hot loop.
        atomicMin(&sBmin[nt + l15], __float_as_uint(colmin));   // ds_min_u32
    }

    // ---- forward: min over the 16 N-columns held across lanes of each half ----
#pragma unroll
    for (int off = 1; off < 16; off <<= 1) {
#pragma unroll
        for (int r = 0; r < 8; ++r)
            fmacc[r] = fminf(fmacc[r], __shfl_xor(fmacc[r], off, 32));
    }
    if (l15 == 0) {
#pragma unroll
        for (int r = 0; r < 8; ++r)
            fmin[b * M_ + m0 + wave * 16 + r + 8 * hi] = fmacc[r];
    }

    __syncthreads();
    // ---- flush per-workgroup column mins to global (one atomic per column) ----
    for (int n = tid; n < N_; n += 256)
        atomicMin(&bming[b * N_ + n], sBmin[n]);             // global_atomic_min_u32
}

// ---------------------------------------------------------------------------
// Kernel 3: final reduction. mean(sqrt(fmin)) + mean(sqrt(bmin)) -> out[0..2]
// ---------------------------------------------------------------------------
__global__ __launch_bounds__(256) void chamfer_reduce_kernel(
    const float* __restrict__ fmin, const unsigned* __restrict__ bmin,
    float* __restrict__ out) {
    __shared__ double sf[256];
    __shared__ double sb[256];
    const int tid = threadIdx.x;
    double fs = 0.0, bs = 0.0;
    for (int i = tid; i < B_ * M_; i += 256) fs += (double)sqrtf(fmin[i]);
    for (int i = tid; i < B_ * N_; i += 256) bs += (double)sqrtf(__uint_as_float(bmin[i]));
    sf[tid] = fs;
    sb[tid] = bs;
    __syncthreads();
    for (int s = 128; s > 0; s >>= 1) {
        if (tid < s) { sf[tid] += sf[tid + s]; sb[tid] += sb[tid + s]; }
        __syncthreads();
    }
    if (tid == 0) {
        float r = (float)(sf[0] / (double)(B_ * M_) + sb[0] / (double)(B_ * N_));
        out[0] = r;
        out[1] = r;
        out[2] = r;
    }
}

// ---------------------------------------------------------------------------
extern "C" void kernel_launch(void* const* d_in, const int* in_sizes, int n_in,
                              void* d_out, int out_size, void* d_ws, size_t ws_size,
                              hipStream_t stream) {
    (void)in_sizes; (void)n_in; (void)out_size; (void)ws_size;
    const float* src = (const float*)d_in[0];   // [B,3,M]
    const float* dst = (const float*)d_in[1];   // [B,3,N]

    // workspace: s2[B*M] | d2[B*N] | fmin[B*M] | bmin[B*N]  (512 KB total)
    float*    s2   = (float*)d_ws;
    float*    d2   = s2 + B_ * M_;
    float*    fmin = d2 + B_ * N_;
    unsigned* bmin = (unsigned*)(fmin + B_ * M_);

    chamfer_init_kernel<<<(B_ * M_ + 255) / 256, 256, 0, stream>>>(
        src, dst, s2, d2, bmin);

    dim3 grid(M_ / 128, B_);
    chamfer_tile_kernel<<<grid, 256, 0, stream>>>(src, dst, s2, d2, fmin, bmin);

    chamfer_reduce_kernel<<<1, 256, 0, stream>>>(fmin, bmin, (float*)d_out);
}